// e2edro_64974265254413
// MI455X (gfx1250) — compile-verified
//
#include <hip/hip_runtime.h>
#include <hip/hip_bf16.h>
#include <math.h>

// ---------------------------------------------------------------------------
// e2e DRO solver for MI455X (gfx1250, wave32, WMMA).
//
// Shapes: X(256x64) Y(256x128) W(128x64) b(128); out = [Z*(256x128), Yhat(256x128)]
//
// Roofline: ~5 GFLOP total, 128KB working set (L2-resident) -> latency bound.
// Strategy: batch 16 scenarios per workgroup so both per-iteration mat-vecs
// become fp32 GEMMs lowered to V_WMMA_F32_16X16X4_F32; keep all solver state
// (z, c, eta, lam, Q) in LDS; per-scenario simplex projection is a wave-local
// bitonic sort in LDS synchronized with s_wait_dscnt (no workgroup barrier).
// ---------------------------------------------------------------------------

typedef __attribute__((ext_vector_type(2))) float v2f;
typedef __attribute__((ext_vector_type(8))) float v8f;

#define N_OBS 256
#define N_X   64
#define N_Y   128
#define NB    16      // scenarios per workgroup (one WMMA N-tile)
#define NITER 150
#define LR0   0.05f

// ===========================================================================
// Kernel 1: Yhat = X @ W^T + b   and   ep = Y - Yhat
// One wave per 16x16 output tile; K=64 -> 16 v_wmma_f32_16x16x4_f32 per wave.
// A-frag (16x4 f32): lane row=l%16, half=l>>4 holds K = 2*half + {0,1}.
// B-frag (4x16):     lane col=l%16 holds B[2*half+{0,1}][col], B[k][n]=W[n][k].
// C/D: vgpr i, lanes 0-15 -> (M=i, N=l); lanes 16-31 -> (M=8+i, N=l-16).
// ===========================================================================
__global__ __launch_bounds__(32) void e2edro_yhat_ep_wmma(
    const float* __restrict__ X, const float* __restrict__ Y,
    const float* __restrict__ W, const float* __restrict__ bvec,
    float* __restrict__ Yhat, float* __restrict__ ep)
{
    const int tile = blockIdx.x;          // 16 M-tiles x 8 N-tiles = 128
    const int tm   = (tile >> 3) * 16;    // row base (0..255)
    const int tn   = (tile & 7) * 16;     // col base (0..127)
    const int lane = threadIdx.x & 31;
    const int row  = lane & 15;
    const int half = lane >> 4;

    v8f acc = {};
    for (int k0 = 0; k0 < N_X; k0 += 4) {
        const int k = k0 + 2 * half;
        v2f a  = *(const v2f*)(X + (tm + row) * N_X + k);  // A[m][k] = X[tm+m][k]
        v2f bf = *(const v2f*)(W + (tn + row) * N_X + k);  // B[k][n] = W[tn+n][k]
        acc = __builtin_amdgcn_wmma_f32_16x16x4_f32(
            false, a, false, bf, (short)0, acc, false, false);
    }

    const int col  = tn + row;
    const float bi = bvec[col];
    #pragma unroll
    for (int i = 0; i < 8; ++i) {
        const int r = tm + half * 8 + i;
        const float yh = acc[i] + bi;
        Yhat[r * N_Y + col] = yh;
        ep[r * N_Y + col]   = Y[r * N_Y + col] - yh;
    }
}

// ===========================================================================
// Kernel 2: 16 DRO solves per workgroup, 150 unrolled subgradient iterations.
// ===========================================================================
__global__ __launch_bounds__(512) void e2edro_solve_wmma(
    const float* __restrict__ ep, const float* __restrict__ Yhat,
    const float* __restrict__ pDelta, const float* __restrict__ pGamma,
    float* __restrict__ Zout)
{
    __shared__ float Zb[N_Y * NB];        // z, layout [k][s]  (B-frag for phase 1)
    __shared__ float Qs[N_OBS * NB];      // q = active*2r/T, layout [t][s]
    __shared__ float vbuf[NB * N_Y];      // pre-projection v, layout [s][m]
    __shared__ float sbuf[NB * N_Y];      // sort scratch,    layout [s][m]
    __shared__ float pq[16 * 32];         // per-wave partial sums of q
    __shared__ float pa[16 * 32];         // per-wave partial sums of active
    __shared__ float cS[NB], etaS[NB], lamS[NB], thetaS[NB];

    const int tid  = threadIdx.x;
    const int w    = tid >> 5;            // wave id 0..15
    const int lane = tid & 31;
    const int row  = lane & 15;           // WMMA row/col index
    const int half = lane >> 4;           // WMMA K-half
    const int s0   = blockIdx.x * NB;     // first scenario of this block
    const float delta = pDelta[0];
    const float gamma = pGamma[0];

    // init: z = 1/n_y, c = eta = 0, lam = 0.1
    for (int idx = tid; idx < N_Y * NB; idx += 512) Zb[idx] = 1.0f / 128.0f;
    if (tid < NB) { cS[tid] = 0.0f; etaS[tid] = 0.0f; lamS[tid] = 0.1f; }
    __syncthreads();

    for (int it = 0; it < NITER; ++it) {
        const float lr = LR0 / sqrtf(1.0f + (float)it);

        // ---- Phase 1: R = ep @ Z16  (256x128 * 128x16), 16 waves x 32 wmma
        v8f acc = {};
        for (int k0 = 0; k0 < N_Y; k0 += 4) {
            const int k = k0 + 2 * half;
            v2f a = *(const v2f*)(ep + (w * 16 + row) * N_Y + k); // A[t][k]=ep[t][k]
            v2f bf; bf[0] = Zb[k * NB + row]; bf[1] = Zb[(k + 1) * NB + row];
            acc = __builtin_amdgcn_wmma_f32_16x16x4_f32(
                false, a, false, bf, (short)0, acc, false, false);
        }

        // ---- Phase 2a: residual masks + Q + partial reductions
        const int s = row;                 // this lane's scenario column
        const float c  = cS[s];
        const float et = etaS[s];
        const float la = lamS[s];
        float sq = 0.0f, sa = 0.0f;
        #pragma unroll
        for (int i = 0; i < 8; ++i) {
            const int t  = w * 16 + half * 8 + i;
            const float r  = acc[i] - c;
            const float d  = r * r - et + la;          // (r^2-eta) - (-lam)
            const float ac = d > 0.0f ? 1.0f : (d == 0.0f ? 0.5f : 0.0f);
            const float q  = ac * 2.0f * r * (1.0f / (float)N_OBS);
            Qs[t * NB + s] = q;
            sq += q; sa += ac;
        }
        pq[w * 32 + lane] = sq;
        pa[w * 32 + lane] = sa;
        __syncthreads();

        // ---- Phase 2b: deterministic scalar-gradient reduction + updates
        if (tid < NB) {
            float Sq = 0.0f, Sa = 0.0f;
            for (int ww = 0; ww < 16; ++ww) {
                Sq += pq[ww * 32 + tid] + pq[ww * 32 + tid + 16];
                Sa += pa[ww * 32 + tid] + pa[ww * 32 + tid + 16];
            }
            const float g_c   = -Sq;
            const float g_eta = 1.0f - Sa * (1.0f / (float)N_OBS);
            const float g_lam = delta - 1.0f + Sa * (1.0f / (float)N_OBS);
            cS[tid]   -= lr * g_c;
            etaS[tid] -= lr * g_eta;
            const float nl = lamS[tid] - lr * g_lam;
            lamS[tid] = nl > 0.0f ? nl : 0.0f;
        }
        __syncthreads();

        // ---- Phase 3: G = ep^T @ Q (128x256 * 256x16), waves 0..7 x 64 wmma
        if (w < 8) {
            v8f g = {};
            for (int k0 = 0; k0 < N_OBS; k0 += 4) {
                const int k = k0 + 2 * half;
                v2f a;                       // A[m][k] = ep^T[m][k] = ep[k][m]
                a[0] = ep[k * N_Y + (w * 16 + row)];
                a[1] = ep[(k + 1) * N_Y + (w * 16 + row)];
                v2f bf; bf[0] = Qs[k * NB + row]; bf[1] = Qs[(k + 1) * NB + row];
                g = __builtin_amdgcn_wmma_f32_16x16x4_f32(
                    false, a, false, bf, (short)0, g, false, false);
            }
            #pragma unroll
            for (int i = 0; i < 8; ++i) {
                const int m = w * 16 + half * 8 + i;
                const float gz = g[i] - gamma * Yhat[(s0 + s) * N_Y + m];
                const float v  = Zb[m * NB + s] - lr * gz;
                vbuf[s * N_Y + m] = v;
                sbuf[s * N_Y + m] = v;
            }
        }
        __syncthreads();

        // ---- Phase 4: per-wave descending bitonic sort of scenario w's 128
        // candidates (wave-lockstep; LDS same-wave ordering + s_wait_dscnt).
        {
            float* rowp = sbuf + w * N_Y;
            for (int kk = 2; kk <= 128; kk <<= 1) {
                for (int j = kk >> 1; j > 0; j >>= 1) {
                    #pragma unroll
                    for (int uu = 0; uu < 2; ++uu) {
                        const int u = lane + uu * 32;           // 64 pairs/stage
                        const int i = ((u & ~(j - 1)) << 1) | (u & (j - 1));
                        const int l = i | j;
                        const float x = rowp[i];
                        const float y = rowp[l];
                        const bool desc = ((i & kk) == 0);
                        const bool sw = desc ? (x < y) : (x > y);
                        if (sw) { rowp[i] = y; rowp[l] = x; }
                    }
                    asm volatile("s_wait_dscnt 0x0" ::: "memory");
                }
            }
            // exact simplex threshold: rho = #{i : u_i + (1-css_i)/(i+1) > 0},
            // theta = (css[rho-1]-1)/rho   (matches reference semantics)
            if (lane == 0) {
                float css = 0.0f; int cnt = 0;
                for (int i2 = 0; i2 < 128; ++i2) {
                    const float u = rowp[i2];
                    css += u;
                    if (u + (1.0f - css) / (float)(i2 + 1) > 0.0f) cnt++;
                }
                float css2 = 0.0f;
                for (int i2 = 0; i2 < cnt; ++i2) css2 += rowp[i2];
                thetaS[w] = (css2 - 1.0f) / (float)cnt;
            }
        }
        __syncthreads();

        // ---- Phase 5: z = max(v - theta, 0), back into WMMA B-frag layout
        for (int idx = tid; idx < N_Y * NB; idx += 512) {
            const int m = idx >> 4, ss = idx & 15;
            const float v = vbuf[ss * N_Y + m] - thetaS[ss];
            Zb[idx] = v > 0.0f ? v : 0.0f;
        }
        __syncthreads();
    }

    // write Z* for this block's 16 scenarios
    for (int idx = tid; idx < N_Y * NB; idx += 512) {
        const int m = idx >> 4, ss = idx & 15;
        Zout[(s0 + ss) * N_Y + m] = Zb[idx];
    }
}

// ===========================================================================
extern "C" void kernel_launch(void* const* d_in, const int* in_sizes, int n_in,
                              void* d_out, int out_size, void* d_ws, size_t ws_size,
                              hipStream_t stream) {
    const float* X     = (const float*)d_in[0];   // 256*64
    const float* Y     = (const float*)d_in[1];   // 256*128
    const float* W     = (const float*)d_in[2];   // 128*64
    const float* bvec  = (const float*)d_in[3];   // 128
    const float* delta = (const float*)d_in[4];   // 1
    const float* gamma = (const float*)d_in[5];   // 1

    float* out  = (float*)d_out;
    float* Z    = out;                    // Z_star: 256*128
    float* Yhat = out + N_OBS * N_Y;      // Y_hat : 256*128
    float* ep   = (float*)d_ws;           // 256*128 residuals (128 KB scratch)

    e2edro_yhat_ep_wmma<<<128, 32, 0, stream>>>(X, Y, W, bvec, Yhat, ep);
    e2edro_solve_wmma<<<16, 512, 0, stream>>>(ep, Yhat, delta, gamma, Z);
}